// HeteroGAT_dgl_17119739641943
// MI455X (gfx1250) — compile-verified
//
#include <hip/hip_runtime.h>
#include <hip/hip_bf16.h>
#include <math.h>

// ---------------- problem constants (match reference) ----------------
#define NN    50000     // nodes
#define RR    3         // relations
#define EE    250000    // edges per relation
#define PP    100000    // negative pairs
#define FIN   128
#define HH    256
#define SLOPE 0.2f

typedef float v2f __attribute__((ext_vector_type(2)));
typedef float v8f __attribute__((ext_vector_type(8)));
typedef unsigned int u32x4 __attribute__((ext_vector_type(4)));
typedef int i32x4 __attribute__((ext_vector_type(4)));
typedef int i32x8 __attribute__((ext_vector_type(8)));

// ---------------- TDM 2D tile load: global -> LDS (CDNA5 D# descriptor) ----
// D# per ISA ch.8: group0 = {count|flags, lds_addr, global_addr lo, global_addr
// hi | type=2}; group1 packs data_size / pad control / tensor dims / tile dims /
// dim0 stride. Optional groups zero (2D tensor). Tracked by TENSORcnt.
// This toolchain exposes the 6-arg builtin (clang-23 / therock-10.0 form).
__device__ __forceinline__ void tdm_load_2d(unsigned ldsAddr, const void* gaddr,
                                            unsigned td0, unsigned td1,
                                            unsigned tile0, unsigned tile1,
                                            unsigned long stride0,
                                            unsigned padIntervalCode,
                                            unsigned padAmountCode,
                                            bool padEn) {
    unsigned long ga = (unsigned long)(uintptr_t)gaddr;
    u32x4 g0;
    g0.x = 1u;                                             // count=1, user mode
    g0.y = ldsAddr;                                        // LDS byte address
    g0.z = (unsigned)(ga & 0xFFFFFFFFu);                   // global addr [31:0]
    g0.w = (unsigned)((ga >> 32) & 0x01FFFFFFu) | (2u << 30); // [56:32] | type=2

    unsigned w0 = (2u << 16);                              // data_size = 4B
    if (padEn) w0 |= (1u << 20) | (padIntervalCode << 22) | (padAmountCode << 25);
    i32x8 g1;
    g1[0] = (int)w0;
    g1[1] = (int)((td0 & 0xFFFFu) << 16);                  // tensor_dim0[15:0]
    g1[2] = (int)((td0 >> 16) | ((td1 & 0xFFFFu) << 16));  // td0 hi | td1 lo
    g1[3] = (int)((td1 >> 16) | ((tile0 & 0xFFFFu) << 16));// td1 hi | tile_dim0
    g1[4] = (int)(tile1 & 0xFFFFu);                        // tile_dim1, tile_dim2=0
    g1[5] = (int)(unsigned)(stride0 & 0xFFFFFFFFu);        // dim0 stride lo
    g1[6] = (int)(unsigned)((stride0 >> 32) & 0xFFFFu);    // dim0 stride hi
    g1[7] = 0;                                             // dim1 stride unused (2D)

    i32x4 z4 = {0, 0, 0, 0};
    i32x8 z8 = {0, 0, 0, 0, 0, 0, 0, 0};
    __builtin_amdgcn_tensor_load_to_lds(g0, g1, z4, z4, z8, 0);
}

__device__ __forceinline__ unsigned lds_offset(const void* p) {
    // flat shared-memory pointers carry the LDS byte offset in the low 32 bits
    return (unsigned)(uintptr_t)p;
}

// ---------------- WMMA f32 GEMM with TDM-staged LDS tiles ------------------
// Block = 4 waves computing a 32x32 tile (2m x 2n 16x16 wave tiles).
// A tile (32 x K, rows padded by 1 dword in LDS) and B tile (K x 32) are
// DMA'd by the Tensor Data Mover; waves consume them with V_WMMA_F32_16X16X4.
// A frag (16x4 f32): lane<16 -> M=lane, K={k,k+1}; lane>=16 -> K={k+2,k+3}
// B frag (4x16 f32): lane<16 -> K={k,k+1}, N=lane; lane>=16 -> K={k+2,k+3}
// C/D (16x16 f32, 8 VGPRs): c[i]: row = i + 8*(lane>=16), col = lane&15
template <int K, int Nc>
__global__ void gemm_wmma_tdm(const float* __restrict__ A,
                              const float* __restrict__ B,
                              float* __restrict__ C, int M) {
    constexpr int PADK = K + 1;                  // TDM pad: 1 dword per row
    __shared__ float sA[32 * PADK];
    __shared__ float sB[K * 32];

    const int lane  = threadIdx.x & 31;
    const int wave  = threadIdx.x >> 5;
    const int wm    = wave >> 1;                 // 0..1 (m sub-tile)
    const int wn    = wave & 1;                  // 0..1 (n sub-tile)
    const int n0    = blockIdx.x * 32;
    const int mBase = blockIdx.y * 32;

    if (wave == 0) {
        unsigned remRows = (unsigned)(M - mBase); // OOB rows read back zero
        // A tile: 32 rows x K, row-major, LDS rows padded to K+1 dwords
        tdm_load_2d(lds_offset(sA), A + (size_t)mBase * K,
                    (unsigned)K, remRows, (unsigned)K, 32u,
                    (unsigned long)K,
                    (K == 128 ? 6u : 7u) /* pad every K dwords */,
                    0u /* +1 dword */, true);
        // B tile: K rows x 32 cols out of K x Nc
        tdm_load_2d(lds_offset(sB), B + n0,
                    (unsigned)(Nc - n0), (unsigned)K, 32u, (unsigned)K,
                    (unsigned long)Nc, 0u, 0u, false);
        __builtin_amdgcn_s_wait_tensorcnt(0);
    }
    __syncthreads();

    const int half = lane >> 4;                  // 0 or 1
    const int l    = lane & 15;

    v8f c = {0.f, 0.f, 0.f, 0.f, 0.f, 0.f, 0.f, 0.f};
    const float* aRow = sA + (size_t)(wm * 16 + l) * PADK + half * 2;
    const float* bCol = sB + (size_t)(half * 2) * 32 + wn * 16 + l;

    #pragma unroll 8
    for (int k = 0; k < K; k += 4) {
        v2f a, b;
        a.x = aRow[k];
        a.y = aRow[k + 1];
        b.x = bCol[k * 32];
        b.y = bCol[(k + 1) * 32];
        c = __builtin_amdgcn_wmma_f32_16x16x4_f32(
                /*neg_a=*/false, a, /*neg_b=*/false, b,
                /*c_mod=*/(short)0, c, /*reuse_a=*/false, /*reuse_b=*/false);
    }

    const int m0 = mBase + wm * 16;
    if (m0 < M) {                                // wave-uniform guard
        float* Crow = C + (size_t)(m0 + half * 8) * Nc + n0 + wn * 16 + l;
        #pragma unroll
        for (int i = 0; i < 8; ++i) Crow[(size_t)i * Nc] = c[i];
    }
}

// ---------------- el/er = feat . al / feat . ar (wave per node) ----------------
__global__ void scores_kernel(const float* __restrict__ feat,
                              const float* __restrict__ al,
                              const float* __restrict__ ar,
                              float* __restrict__ el, float* __restrict__ er,
                              int nNodes, int Hdim) {
    int gid  = blockIdx.x * blockDim.x + threadIdx.x;
    int node = gid >> 5;
    int lane = gid & 31;
    if (node >= nNodes) return;
    const float* f = feat + (size_t)node * Hdim;
    float sl = 0.f, sr = 0.f;
    for (int j = lane; j < Hdim; j += 32) {
        float v = f[j];
        sl += v * al[j];
        sr += v * ar[j];
    }
    for (int o = 16; o > 0; o >>= 1) {
        sl += __shfl_down(sl, o, 32);
        sr += __shfl_down(sr, o, 32);
    }
    if (lane == 0) { el[node] = sl; er[node] = sr; }
}

// ---------------- float atomic max via signed/unsigned int trick ----------------
__device__ __forceinline__ void atomicMaxFloat(float* addr, float v) {
    if (v >= 0.f)
        atomicMax((int*)addr, __float_as_int(v));
    else
        atomicMin((unsigned int*)addr, __float_as_uint(v));
}

__device__ __forceinline__ float leaky(float v) {
    return v > 0.f ? v : v * SLOPE;
}

__global__ void edge_max_kernel(const float* __restrict__ el,
                                const float* __restrict__ er,
                                const int* __restrict__ src,
                                const int* __restrict__ dst,
                                float* __restrict__ m, int nEdges) {
    int e = blockIdx.x * blockDim.x + threadIdx.x;
    if (e >= nEdges) return;
    float v = leaky(el[src[e]] + er[dst[e]]);
    atomicMaxFloat(&m[dst[e]], v);
}

__global__ void edge_a_kernel(const float* __restrict__ el,
                              const float* __restrict__ er,
                              const int* __restrict__ src,
                              const int* __restrict__ dst,
                              const float* __restrict__ m,
                              float* __restrict__ denom,
                              float* __restrict__ aw, int nEdges) {
    int e = blockIdx.x * blockDim.x + threadIdx.x;
    if (e >= nEdges) return;
    int d = dst[e];
    float v = leaky(el[src[e]] + er[d]);
    float a = expf(v - m[d]);
    aw[e] = a;
    atomicAdd(&denom[d], a);
}

// wave per edge: out[dst] += feat[src] * a   (atomics resolve in L2; 51MB << 192MB L2)
__global__ void scatter_kernel(const float* __restrict__ feat,
                               const float* __restrict__ aw,
                               const int* __restrict__ src,
                               const int* __restrict__ dst,
                               float* __restrict__ out,
                               int nEdges, int Hdim) {
    int gid  = blockIdx.x * blockDim.x + threadIdx.x;
    int e    = gid >> 5;
    int lane = gid & 31;
    if (e >= nEdges) return;
    float a = aw[e];
    const float* fs = feat + (size_t)src[e] * Hdim;
    float* od = out + (size_t)dst[e] * Hdim;
    for (int j = lane * 4; j < Hdim; j += 32 * 4) {
        float4 v = *(const float4*)(fs + j);
        atomicAdd(od + j + 0, v.x * a);
        atomicAdd(od + j + 1, v.y * a);
        atomicAdd(od + j + 2, v.z * a);
        atomicAdd(od + j + 3, v.w * a);
    }
}

// acc += out_r / max(denom,1e-9) + b_r   (per element)
__global__ void accum_kernel(const float* __restrict__ out_r,
                             const float* __restrict__ denom,
                             const float* __restrict__ b,
                             float* __restrict__ acc,
                             int nNodes, int Hdim) {
    size_t i = (size_t)blockIdx.x * blockDim.x + threadIdx.x;
    if (i >= (size_t)nNodes * Hdim) return;
    int n = (int)(i / Hdim);
    int j = (int)(i % Hdim);
    float inv = 1.f / fmaxf(denom[n], 1e-9f);
    acc[i] += out_r[i] * inv + b[j];
}

__global__ void scale_relu_kernel(float* __restrict__ p, size_t n, float s, int doRelu) {
    size_t i = (size_t)blockIdx.x * blockDim.x + threadIdx.x;
    if (i >= n) return;
    float v = p[i] * s;
    p[i] = doRelu ? fmaxf(v, 0.f) : v;
}

__global__ void fill_kernel(float* __restrict__ p, float v, size_t n) {
    size_t i = (size_t)blockIdx.x * blockDim.x + threadIdx.x;
    if (i < n) p[i] = v;
}

// ---------------- final pair scoring (wave per pair) ----------------
__global__ void pair_kernel(const float* __restrict__ hn,
                            const int* __restrict__ edges,   // (R,2,E)
                            const int* __restrict__ n_pairs, // (P,2)
                            const float* __restrict__ Wlin,  // (2*FIN)
                            const float* __restrict__ blin,
                            float* __restrict__ out, int nTotal) {
    int gid  = blockIdx.x * blockDim.x + threadIdx.x;
    int i    = gid >> 5;
    int lane = gid & 31;
    if (i >= nTotal) return;
    int s, d;
    if (i < RR * EE) {
        int r = i / EE, e = i - r * EE;
        const int* eb = edges + (size_t)r * 2 * EE;
        s = eb[e];
        d = eb[EE + e];
    } else {
        int p = i - RR * EE;
        s = n_pairs[2 * p];
        d = n_pairs[2 * p + 1];
    }
    const float* hs = hn + (size_t)s * FIN;
    const float* hd = hn + (size_t)d * FIN;
    float sum = 0.f;
    for (int j = lane; j < FIN; j += 32) {
        sum += fmaxf(hs[j], 0.f) * Wlin[j] + fmaxf(hd[j], 0.f) * Wlin[FIN + j];
    }
    for (int o = 16; o > 0; o >>= 1) sum += __shfl_down(sum, o, 32);
    if (lane == 0) {
        float z = sum + blin[0];
        out[i] = 1.f / (1.f + expf(-z));
    }
}

// ---------------- host-side orchestration ----------------
extern "C" void kernel_launch(void* const* d_in, const int* in_sizes, int n_in,
                              void* d_out, int out_size, void* d_ws, size_t ws_size,
                              hipStream_t stream) {
    const float* x     = (const float*)d_in[0];
    const int*   edges = (const int*)d_in[1];   // (R,2,E)
    const int*   npair = (const int*)d_in[2];   // (P,2)
    const float* W1    = (const float*)d_in[3]; // (R,FIN,H)
    const float* al1   = (const float*)d_in[4]; // (R,H)
    const float* ar1   = (const float*)d_in[5];
    const float* b1    = (const float*)d_in[6];
    const float* W2    = (const float*)d_in[7]; // (R,H,FIN)
    const float* al2   = (const float*)d_in[8]; // (R,FIN)
    const float* ar2   = (const float*)d_in[9];
    const float* b2    = (const float*)d_in[10];
    const float* Wlin  = (const float*)d_in[11]; // (2*FIN)
    const float* blin  = (const float*)d_in[12];
    float* outp = (float*)d_out;

    float* ws = (float*)d_ws;
    // workspace layout (floats)
    size_t offH    = 0;                          // h accumulator  N*H
    size_t offHn   = offH  + (size_t)NN * HH;    // hn accumulator N*FIN
    size_t offEl   = offHn + (size_t)NN * FIN;   // N
    size_t offEr   = offEl + NN;                 // N
    size_t offM    = offEr + NN;                 // N
    size_t offD    = offM  + NN;                 // N
    size_t offAw   = offD  + NN;                 // E
    size_t offFeat = offAw + EE;                 // N*H (reused as N*FIN in layer2)
    size_t offOut  = offFeat + (size_t)NN * HH;  // N*H (reused as N*FIN in layer2)

    float* hAcc  = ws + offH;
    float* hnAcc = ws + offHn;
    float* el    = ws + offEl;
    float* er    = ws + offEr;
    float* mBuf  = ws + offM;
    float* dBuf  = ws + offD;
    float* aw    = ws + offAw;
    float* featR = ws + offFeat;
    float* outR  = ws + offOut;

    dim3 gemmBlock(128);                          // 4 waves, 32x32 block tile
    dim3 gemmGrid1(HH / 32, (NN + 31) / 32);
    dim3 gemmGrid2(FIN / 32, (NN + 31) / 32);

    const int waveBlk   = 256;
    const int nodeWaves = (NN * 32 + waveBlk - 1) / waveBlk;
    const int edgeThr   = (EE + waveBlk - 1) / waveBlk;
    const int edgeWaves = (EE * 32 + waveBlk - 1) / waveBlk;
    const int elemBlk1  = (int)(((size_t)NN * HH + waveBlk - 1) / waveBlk);
    const int elemBlk2  = (int)(((size_t)NN * FIN + waveBlk - 1) / waveBlk);
    const int fillBlkN  = (NN + waveBlk - 1) / waveBlk;

    (void)hipMemsetAsync(hAcc,  0, (size_t)NN * HH  * sizeof(float), stream);
    (void)hipMemsetAsync(hnAcc, 0, (size_t)NN * FIN * sizeof(float), stream);

    // ---------------- layer 1 (input x, F_IN -> H) ----------------
    for (int r = 0; r < RR; ++r) {
        const int* srcE = edges + (size_t)r * 2 * EE;
        const int* dstE = srcE + EE;

        (void)hipMemsetAsync(outR, 0, (size_t)NN * HH * sizeof(float), stream);
        (void)hipMemsetAsync(dBuf, 0, (size_t)NN * sizeof(float), stream);
        fill_kernel<<<fillBlkN, waveBlk, 0, stream>>>(mBuf, -INFINITY, NN);

        gemm_wmma_tdm<FIN, HH><<<gemmGrid1, gemmBlock, 0, stream>>>(
            x, W1 + (size_t)r * FIN * HH, featR, NN);
        scores_kernel<<<nodeWaves, waveBlk, 0, stream>>>(
            featR, al1 + (size_t)r * HH, ar1 + (size_t)r * HH, el, er, NN, HH);
        edge_max_kernel<<<edgeThr, waveBlk, 0, stream>>>(el, er, srcE, dstE, mBuf, EE);
        edge_a_kernel<<<edgeThr, waveBlk, 0, stream>>>(el, er, srcE, dstE, mBuf, dBuf, aw, EE);
        scatter_kernel<<<edgeWaves, waveBlk, 0, stream>>>(featR, aw, srcE, dstE, outR, EE, HH);
        accum_kernel<<<elemBlk1, waveBlk, 0, stream>>>(outR, dBuf, b1 + (size_t)r * HH, hAcc, NN, HH);
    }
    // h = relu(mean over R)
    scale_relu_kernel<<<elemBlk1, waveBlk, 0, stream>>>(hAcc, (size_t)NN * HH, 1.f / RR, 1);

    // ---------------- layer 2 (input h, H -> F_IN) ----------------
    for (int r = 0; r < RR; ++r) {
        const int* srcE = edges + (size_t)r * 2 * EE;
        const int* dstE = srcE + EE;

        (void)hipMemsetAsync(outR, 0, (size_t)NN * FIN * sizeof(float), stream);
        (void)hipMemsetAsync(dBuf, 0, (size_t)NN * sizeof(float), stream);
        fill_kernel<<<fillBlkN, waveBlk, 0, stream>>>(mBuf, -INFINITY, NN);

        gemm_wmma_tdm<HH, FIN><<<gemmGrid2, gemmBlock, 0, stream>>>(
            hAcc, W2 + (size_t)r * HH * FIN, featR, NN);
        scores_kernel<<<nodeWaves, waveBlk, 0, stream>>>(
            featR, al2 + (size_t)r * FIN, ar2 + (size_t)r * FIN, el, er, NN, FIN);
        edge_max_kernel<<<edgeThr, waveBlk, 0, stream>>>(el, er, srcE, dstE, mBuf, EE);
        edge_a_kernel<<<edgeThr, waveBlk, 0, stream>>>(el, er, srcE, dstE, mBuf, dBuf, aw, EE);
        scatter_kernel<<<edgeWaves, waveBlk, 0, stream>>>(featR, aw, srcE, dstE, outR, EE, FIN);
        accum_kernel<<<elemBlk2, waveBlk, 0, stream>>>(outR, dBuf, b2 + (size_t)r * FIN, hnAcc, NN, FIN);
    }
    // hn = mean over R (no relu here; relu applied per-pair)
    scale_relu_kernel<<<elemBlk2, waveBlk, 0, stream>>>(hnAcc, (size_t)NN * FIN, 1.f / RR, 0);

    // ---------------- pair scoring ----------------
    const int nTotal = RR * EE + PP;   // 850000
    const int pairBlocks = (nTotal * 32 + waveBlk - 1) / waveBlk;
    pair_kernel<<<pairBlocks, waveBlk, 0, stream>>>(hnAcc, edges, npair, Wlin, blin, outp, nTotal);
}